// TopologicalLoss_15015205667337
// MI455X (gfx1250) — compile-verified
//
#include <hip/hip_runtime.h>

// ---------------- problem constants (from reference) ----------------
#define B_SAMP   8
#define N_CH     4
#define HW       (1024 * 1024)      // H*W
#define HW4      (HW / 4)           // float4 groups per [b, ch] plane = 262144 = 2^18
#define NPIX     (B_SAMP * HW)      // 8,388,608
#define GROUPS   (NPIX / 4)         // 2,097,152 float4 pixel-groups
#define BLOCKS1  2048
#define THREADS1 256
#define GSTRIDE  (BLOCKS1 * THREADS1)   // 524,288 -> exactly 4 iterations/thread

typedef __attribute__((ext_vector_type(2))) float v2f;
typedef __attribute__((ext_vector_type(8))) float v8f;

// ---------------------------------------------------------------
// Full 32-lane wave sum using V_WMMA_F32_16X16X4_F32.
// Stage 1: A[m,k] = (k==0 ? acc_m : k==2 ? acc_{m+16} : 0), B = ones(4x16)
//          -> D[m,n] = acc_m + acc_{m+16}   (constant across n)
// Stage 2: A = ones(16x4), B = pairs of D's VGPRs (rows {j,j+1,j+8,j+9}),
//          chained through C over j = 0,2,4,6 -> every element = wave total.
// Requires EXEC == all ones (call from fully-converged code only).
// ---------------------------------------------------------------
__device__ __forceinline__ float wave_reduce_wmma(float acc) {
  v2f a;   a[0] = acc;  a[1] = 0.0f;   // lanes 0-15: K=0; lanes 16-31: K=2
  v2f one; one[0] = 1.0f; one[1] = 1.0f;
  v8f c = {};
  v8f d = __builtin_amdgcn_wmma_f32_16x16x4_f32(
      /*neg_a=*/false, a, /*neg_b=*/false, one,
      /*c_mod=*/(short)0, c, /*reuse_a=*/false, /*reuse_b=*/false);

  v8f s = {};
#pragma unroll
  for (int j = 0; j < 8; j += 2) {
    v2f bcol; bcol[0] = d[j]; bcol[1] = d[j + 1];
    s = __builtin_amdgcn_wmma_f32_16x16x4_f32(
        false, one, false, bcol, (short)0, s, false, false);
  }
  return s[0];   // every lane/element now holds the 32-lane sum
}

// ---------------------------------------------------------------
// Kernel 1: streaming partial reduction.
//   acc += (0.25*(c0+c1+c2+c3) * w - r)^2  over all pixels.
// One float4 group per iteration: 4 channel float4 loads + int4 w + int4 r.
// ---------------------------------------------------------------
__global__ void __launch_bounds__(THREADS1)
topo_partial_kernel(const float4* __restrict__ lh4,
                    const int4*   __restrict__ w4,
                    const int4*   __restrict__ r4,
                    float*        __restrict__ partials) {
  const int tid = blockIdx.x * THREADS1 + threadIdx.x;
  float acc = 0.0f;

#pragma unroll
  for (int it = 0; it < GROUPS / GSTRIDE; ++it) {
    const int g   = tid + it * GSTRIDE;       // global float4-group index
    const int b   = g >> 18;                  // g / HW4
    const int hw4 = g & (HW4 - 1);            // g % HW4
    const float4* base = lh4 + (size_t)(b * N_CH) * HW4 + hw4;  // channel 0

    const float4 c0 = base[0];
    const float4 c1 = base[1 * HW4];
    const float4 c2 = base[2 * HW4];
    const float4 c3 = base[3 * HW4];
    const int4   wv = w4[g];
    const int4   rv = r4[g];

    float m, t;
    m = (c0.x + c1.x + c2.x + c3.x) * 0.25f;
    t = m * (float)wv.x - (float)rv.x;  acc = fmaf(t, t, acc);
    m = (c0.y + c1.y + c2.y + c3.y) * 0.25f;
    t = m * (float)wv.y - (float)rv.y;  acc = fmaf(t, t, acc);
    m = (c0.z + c1.z + c2.z + c3.z) * 0.25f;
    t = m * (float)wv.z - (float)rv.z;  acc = fmaf(t, t, acc);
    m = (c0.w + c1.w + c2.w + c3.w) * 0.25f;
    t = m * (float)wv.w - (float)rv.w;  acc = fmaf(t, t, acc);
  }

  // wave32 reduction via WMMA, then cross-wave combine in LDS
  const float wsum = wave_reduce_wmma(acc);

  __shared__ float smem[THREADS1 / 32];
  const int lane = threadIdx.x & 31;
  const int wid  = threadIdx.x >> 5;
  if (lane == 0) smem[wid] = wsum;
  __syncthreads();

  if (threadIdx.x == 0) {
    float t = 0.0f;
#pragma unroll
    for (int i = 0; i < THREADS1 / 32; ++i) t += smem[i];
    partials[blockIdx.x] = t;
  }
}

// ---------------------------------------------------------------
// Kernel 2: reduce the 2048 block partials, write loss = total / B.
// ---------------------------------------------------------------
__global__ void __launch_bounds__(256)
topo_final_kernel(const float* __restrict__ partials,
                  float*       __restrict__ out) {
  float acc = 0.0f;
#pragma unroll
  for (int i = threadIdx.x; i < BLOCKS1; i += 256) acc += partials[i];

  const float wsum = wave_reduce_wmma(acc);

  __shared__ float smem[8];
  const int lane = threadIdx.x & 31;
  const int wid  = threadIdx.x >> 5;
  if (lane == 0) smem[wid] = wsum;
  __syncthreads();

  if (threadIdx.x == 0) {
    float t = 0.0f;
#pragma unroll
    for (int i = 0; i < 8; ++i) t += smem[i];
    out[0] = t * (1.0f / (float)B_SAMP);
  }
}

// ---------------------------------------------------------------
// Harness entry point.
// d_in[0]=likelihood f32 [8,4,1024,1024], d_in[1]=gt (UNUSED by reference),
// d_in[2]=weight_map i32 [8,1024,1024],   d_in[3]=ref_map i32 [8,1024,1024]
// d_out: 1 float. d_ws: >= 2048 floats used for block partials.
// ---------------------------------------------------------------
extern "C" void kernel_launch(void* const* d_in, const int* in_sizes, int n_in,
                              void* d_out, int out_size, void* d_ws, size_t ws_size,
                              hipStream_t stream) {
  (void)in_sizes; (void)n_in; (void)out_size; (void)ws_size;
  const float4* lh4 = (const float4*)d_in[0];
  const int4*   w4  = (const int4*)d_in[2];
  const int4*   r4  = (const int4*)d_in[3];
  float* partials = (float*)d_ws;
  float* out      = (float*)d_out;

  topo_partial_kernel<<<BLOCKS1, THREADS1, 0, stream>>>(lh4, w4, r4, partials);
  topo_final_kernel<<<1, 256, 0, stream>>>(partials, out);
}